// CausalGraphLearner_39822936769259
// MI455X (gfx1250) — compile-verified
//
#include <hip/hip_runtime.h>
#include <hip/hip_bf16.h>
#include <math.h>

typedef float v2f __attribute__((ext_vector_type(2)));
typedef float v8f __attribute__((ext_vector_type(8)));

#define HD 256          // hidden dim H
#define NV 64           // number of variables V
#define NB 32           // batch B
#define M_TOTAL (NB * NV)   // 2048 rows in all GEMMs
#define A_PAD 68        // 64 + 4 floats: stride%64banks = 4 -> conflict-free

// ---------------------------------------------------------------------------
// Kernel 1: adjacency = sigmoid(logits) with zeroed diagonal, plus column sums
// colsum[i] = sum_j adj[j,i]   (needed to fold b2 into the final GEMM epilogue)
// ---------------------------------------------------------------------------
__global__ void adj_kernel(const float* __restrict__ logits,
                           float* __restrict__ adj_out,
                           float* __restrict__ colsum) {
    const int i = blockIdx.x;   // column
    const int j = threadIdx.x;  // row
    float x = logits[j * NV + i];
    float a = 1.0f / (1.0f + __expf(-x));
    if (j == i) a = 0.0f;
    adj_out[j * NV + i] = a;

    __shared__ float red[NV];
    red[j] = a;
    __syncthreads();
    for (int s = NV / 2; s > 0; s >>= 1) {
        if (j < s) red[j] += red[j + s];
        __syncthreads();
    }
    if (j == 0) colsum[i] = red[0];
}

// ---------------------------------------------------------------------------
// WMMA f32 GEMM: OUT[M_TOTAL x 256] = X[M_TOTAL x 256] @ W[256 x 256] (row-major)
// Block = 128 threads (4 waves, wave32). Each wave -> one 16x16 tile,
// block -> 64 rows x 16 cols. B tile staged fully in LDS (16KB); A staged in
// 64-wide K chunks (17KB, padded stride for bank-conflict-free ds reads).
// Uses V_WMMA_F32_16X16X4_F32 (exact f32, K-step 4, 64 WMMA per tile).
// add_bias!=0 epilogue: OUT += colsum[row%64] * b2[col]
// ---------------------------------------------------------------------------
__global__ void __launch_bounds__(128)
gemm_wmma_kernel(const float* __restrict__ X, const float* __restrict__ W,
                 float* __restrict__ OUT,
                 const float* __restrict__ colsum, const float* __restrict__ b2,
                 int add_bias) {
    __shared__ float lds_a[64 * A_PAD];   // 17,408 B
    __shared__ float lds_b[HD * 16];      // 16,384 B

    const int lane = threadIdx.x;             // 0..31
    const int w    = threadIdx.y;             // wave id 0..3
    const int tid  = w * 32 + lane;           // 0..127
    const int m_base = blockIdx.x * 64;
    const int n_base = blockIdx.y * 16;

    // Stage full B tile: W[k][n_base+n], k=0..255, n=0..15 (float4-coalesced)
    #pragma unroll
    for (int t = 0; t < 8; ++t) {
        int idx4 = tid + t * 128;             // 1024 float4 total
        int k = idx4 >> 2;
        int n = (idx4 & 3) * 4;
        float4 v = *(const float4*)(W + (size_t)k * HD + n_base + n);
        *(float4*)(&lds_b[k * 16 + n]) = v;
    }

    // Fragment index helpers (ISA 7.12.2, 32-bit layouts)
    const int koff = (lane >> 4) * 2;   // lanes 16..31 hold K+2
    const int mn   = lane & 15;         // row (A) / col (B,D)

    v8f acc = {};

    for (int kc = 0; kc < HD; kc += 64) {
        __syncthreads();   // previous chunk consumed (also fences B stage)
        // Stage A chunk: X[m_base+row][kc + k], row=0..63, k=0..63
        #pragma unroll
        for (int t = 0; t < 8; ++t) {
            int idx4 = tid + t * 128;         // 1024 float4 total
            int row = idx4 >> 4;
            int k = (idx4 & 15) * 4;
            float4 v = *(const float4*)(X + (size_t)(m_base + row) * HD + kc + k);
            *(float4*)(&lds_a[row * A_PAD + k]) = v;
        }
        __syncthreads();

        #pragma unroll
        for (int k0 = 0; k0 < 64; k0 += 4) {
            // A frag: lane holds {X[m][k], X[m][k+1]}, m = 16w+mn, k = k0+koff
            v2f a = *(const v2f*)(&lds_a[(w * 16 + mn) * A_PAD + k0 + koff]);
            // B frag: lane holds {W[k][n], W[k+1][n]}, n = mn
            v2f b;
            b.x = lds_b[(kc + k0 + koff) * 16 + mn];
            b.y = lds_b[(kc + k0 + koff + 1) * 16 + mn];
            acc = __builtin_amdgcn_wmma_f32_16x16x4_f32(
                false, a, false, b, (short)0, acc, false, false);
        }
    }

    // D layout: VGPR r -> row r + 8*(lane>>4), col = lane&15
    const int gcol = n_base + mn;
    const int rbase = m_base + w * 16 + ((lane >> 4) * 8);
    #pragma unroll
    for (int r = 0; r < 8; ++r) {
        int grow = rbase + r;
        float v = acc[r];
        if (add_bias) v += colsum[grow & (NV - 1)] * b2[gcol];
        OUT[(size_t)grow * HD + gcol] = v;
    }
}

// ---------------------------------------------------------------------------
// Kernel 3: G[b,i,h] = sum_j adj[j,i] * relu(C[b,j,h] + E[b,i,h] + b1[h])
// One block per (b,i); 256 threads = h. C[b] rows are read coalesced over h.
// ---------------------------------------------------------------------------
__global__ void __launch_bounds__(HD)
gather_relu_kernel(const float* __restrict__ Cc, const float* __restrict__ Ee,
                   const float* __restrict__ adj, const float* __restrict__ b1,
                   float* __restrict__ G) {
    const int h = threadIdx.x;
    const int i = blockIdx.x & (NV - 1);
    const int b = blockIdx.x >> 6;

    __shared__ float acol[NV];
    if (h < NV) acol[h] = adj[h * NV + i];
    __syncthreads();

    const float e = Ee[((size_t)(b * NV + i)) * HD + h] + b1[h];
    const float* crow = Cc + (size_t)b * NV * HD + h;

    float accu = 0.0f;
    #pragma unroll 8
    for (int j = 0; j < NV; ++j) {
        float t = crow[(size_t)j * HD] + e;
        accu += acol[j] * fmaxf(t, 0.0f);
    }
    G[((size_t)(b * NV + i)) * HD + h] = accu;
}

// ---------------------------------------------------------------------------
// Launch: adjacency+colsum -> h_cause -> h_eff -> weighted-relu-gather -> GEMM
// effects = G @ W2 + colsum[i]*b2   (W2/b2 commuted past the linear j-sum)
// ---------------------------------------------------------------------------
extern "C" void kernel_launch(void* const* d_in, const int* in_sizes, int n_in,
                              void* d_out, int out_size, void* d_ws, size_t ws_size,
                              hipStream_t stream) {
    const float* variables  = (const float*)d_in[0];  // [32,64,256]
    const float* adj_logits = (const float*)d_in[1];  // [64,64]
    const float* W1         = (const float*)d_in[2];  // [512,256]
    const float* b1         = (const float*)d_in[3];  // [256]
    const float* W2         = (const float*)d_in[4];  // [256,256]
    const float* b2         = (const float*)d_in[5];  // [256]

    float* effects   = (float*)d_out;                 // [32,64,256] = 524288
    float* adjacency = effects + (size_t)M_TOTAL * HD; // [64,64] = 4096

    float* ws     = (float*)d_ws;
    float* Cb     = ws;                               // [2048,256] h_cause
    float* Eb     = ws + (size_t)M_TOTAL * HD;        // [2048,256] h_eff
    float* G      = ws + 2 * (size_t)M_TOTAL * HD;    // [2048,256] gathered
    float* colsum = ws + 3 * (size_t)M_TOTAL * HD;    // [64]

    adj_kernel<<<dim3(NV), dim3(NV), 0, stream>>>(adj_logits, adjacency, colsum);

    dim3 ggrid(M_TOTAL / 64, HD / 16);   // (32, 16)
    dim3 gblock(32, 4);                  // 4 waves (wave32)

    // h_cause = variables @ W1[:256], h_eff = variables @ W1[256:]
    gemm_wmma_kernel<<<ggrid, gblock, 0, stream>>>(variables, W1, Cb,
                                                   nullptr, nullptr, 0);
    gemm_wmma_kernel<<<ggrid, gblock, 0, stream>>>(variables, W1 + HD * HD, Eb,
                                                   nullptr, nullptr, 0);

    gather_relu_kernel<<<dim3(M_TOTAL), dim3(HD), 0, stream>>>(Cb, Eb, adjacency,
                                                               b1, G);

    // effects = G @ W2 + colsum[i] * b2
    gemm_wmma_kernel<<<ggrid, gblock, 0, stream>>>(G, W2, effects,
                                                   colsum, b2, 1);
}